// RISNetPI_77197742179013
// MI455X (gfx1250) — compile-verified
//
#include <hip/hip_runtime.h>

// ---------------------------------------------------------------------------
// RISNet-PI fully fused forward for MI455X (gfx1250, wave32, WMMA).
// One workgroup (1024 thr = 32 waves) per batch element; each wave owns
// 4 tiles of 16 positions for all 4 users. fp32 WMMA 16x16x4 for the 1x1
// convs; n-uniform feature channels folded into per-user effective biases.
// Round 7: activation state in 256 KB dynamic LDS (of the WGP's 320 KB);
// sched_barrier(0) after EVERY user iteration stops the scheduler from
// keeping 4 WMMA accumulators live at once -> peak pressure < 128 VGPRs.
// ---------------------------------------------------------------------------

typedef float    v2f __attribute__((ext_vector_type(2)));
typedef float    v8f __attribute__((ext_vector_type(8)));
typedef _Float16 h2  __attribute__((ext_vector_type(2)));

#define UU 4
#define FF 4
#define NN 2048
#define HH 16
#define CINC 36
#define TT 4
#define NWAVES 32
#define NTHREADS 1024
// per-(u,n) record: 16 halves = el ch0..7 (halves 0..7), ol ch0..7 (8..15)
#define REC_HALVES 16
#define FA_BYTES (UU * NN * REC_HALVES * 2)   // 256 KB dynamic LDS

static __device__ __forceinline__ v8f wmma4(v2f a, v2f b, v8f c) {
  // D = A(16x4) * B(4x16) + C, fp32.
  return __builtin_amdgcn_wmma_f32_16x16x4_f32(false, a, false, b,
                                               (short)0, c, false, false);
}

extern __shared__ __align__(16) char dynsmem[];

__launch_bounds__(NTHREADS)
__global__ void risnet_pi_kernel(
    const float* __restrict__ channel,
    const float* __restrict__ We1, const float* __restrict__ be1,
    const float* __restrict__ We,  const float* __restrict__ be,
    const float* __restrict__ Wo1, const float* __restrict__ bo1,
    const float* __restrict__ Wo,  const float* __restrict__ bo,
    const float* __restrict__ W8,  const float* __restrict__ b8,
    float* __restrict__ out)
{
  // per-user n-mean raw sums of relu'd ego / opp (finalized each layer)
  __shared__ float sAccE[UU][HH];
  __shared__ float sAccO[UU][HH];
  // per-wave partials for deterministic tree reduction
  __shared__ float sPartE[NWAVES][UU][HH];
  __shared__ float sPartO[NWAVES][UU][HH];
  // current-layer weights as channel-pairs: [ego|opp][h][c/2]
  __shared__ v2f   sW2[2][HH][CINC / 2];
  // per-user effective bias (uniform eg/og chunks folded in): [ego|opp][u][h]
  __shared__ float sBeff[2][UU][HH];

  _Float16* fAh = (_Float16*)dynsmem;   // [u][n][16 halves]

  const int  b    = blockIdx.x;
  const int  tid  = threadIdx.x;
  const int  lane = tid & 31;
  const int  wave = tid >> 5;
  const int  p    = lane & 15;     // D: channel | A: position | B: output ch
  const bool hiH  = (lane >= 16);
  const int  ksel = hiH ? 2 : 0;   // K element selector for A/B VGPR layout
  const float invN = 1.0f / (float)NN;
  const float inv3 = 1.0f / 3.0f;

  const float* chb = channel + (size_t)b * (UU * FF * NN);

  // Persistent A-format raw-channel operands (feat cols 0..3), f16-packed.
  h2 chA[UU][TT];
  #pragma unroll
  for (int u = 0; u < UU; ++u) {
    #pragma unroll
    for (int t = 0; t < TT; ++t) {
      const int n0 = (wave * TT + t) * 16;
      chA[u][t].x =
          (_Float16)chb[(size_t)(u * FF + ksel) * NN + n0 + p];
      chA[u][t].y =
          (_Float16)chb[(size_t)(u * FF + ksel + 1) * NN + n0 + p];
    }
    __builtin_amdgcn_sched_barrier(0);
  }

  float vE[UU], vO[UU];
  const int cbs[4] = {4, 8, 20, 24};

  // D-format result (lane = channel) stored straight into the LDS state in
  // A-friendly records; per-wave DS ops are in-order -> no barrier needed.
  auto xstore = [&](const v8f& d, int u, int n0, int half0) {
    if (p < 8) {
      _Float16* basep =
          fAh + ((size_t)u * NN + n0 + (hiH ? 8 : 0)) * REC_HALVES + half0 + p;
      #pragma unroll
      for (int k = 0; k < 8; ++k)
        basep[(size_t)k * REC_HALVES] = (_Float16)d[k];
    }
  };

  // deterministic workgroup reduction of the per-layer n-mean sums
  auto reduce_means = [&]() {
    #pragma unroll
    for (int u = 0; u < UU; ++u) {
      float e = vE[u] + __shfl_xor(vE[u], 16, 32);
      float o = vO[u] + __shfl_xor(vO[u], 16, 32);
      if (!hiH) {
        sPartE[wave][u][p] = e;
        sPartO[wave][u][p] = o;
      }
    }
    __syncthreads();
    if (tid < 2 * UU * HH) {
      const int which = tid >> 6;
      const int rem = tid & 63;
      const int u = rem >> 4, c = rem & 15;
      float s = 0.f;
      for (int w = 0; w < NWAVES; ++w)
        s += which ? sPartO[w][u][c] : sPartE[w][u][c];
      if (which) sAccO[u][c] = s; else sAccE[u][c] = s;
    }
    __syncthreads();
  };

  // ======================= layer 0 (Cin = F = 4) ==========================
  {
    const v2f wE0 = { We1[p * FF + ksel], We1[p * FF + ksel + 1] };
    const v2f wO0 = { Wo1[p * FF + ksel], Wo1[p * FF + ksel + 1] };
    const float bE = be1[p], bO = bo1[p];

    #pragma unroll
    for (int u = 0; u < UU; ++u) { vE[u] = 0.f; vO[u] = 0.f; }

    #pragma unroll
    for (int t = 0; t < TT; ++t) {
      const int n0 = (wave * TT + t) * 16;
      auto conv0 = [&](int u, const v2f& w, float bias) -> v8f {
        v8f d;
        #pragma unroll
        for (int k = 0; k < 8; ++k) d[k] = bias;
        v2f a;
        a.x = (float)chA[u][t].x;
        a.y = (float)chA[u][t].y;
        return wmma4(a, w, d);
      };

      v8f os;
      #pragma unroll
      for (int k = 0; k < 8; ++k) os[k] = 0.f;
      #pragma unroll
      for (int u = 0; u < UU; ++u) {
        v8f o = conv0(u, wO0, bO);
        #pragma unroll
        for (int k = 0; k < 8; ++k) {
          o[k] = fmaxf(o[k], 0.f);
          os[k] += o[k];
          vO[u] += o[k];
        }
        __builtin_amdgcn_sched_barrier(0);   // keep 1 accumulator live
      }
      #pragma unroll
      for (int u = 0; u < UU; ++u) {
        v8f e = conv0(u, wE0, bE);
        #pragma unroll
        for (int k = 0; k < 8; ++k) {
          e[k] = fmaxf(e[k], 0.f);
          vE[u] += e[k];
        }
        v8f o = conv0(u, wO0, bO);
        v8f l;
        #pragma unroll
        for (int k = 0; k < 8; ++k) {
          o[k] = fmaxf(o[k], 0.f);
          l[k] = (os[k] - o[k]) * inv3;
        }
        xstore(e, u, n0, 0);   // el -> halves 0..7
        xstore(l, u, n0, 8);   // ol -> halves 8..15
        __builtin_amdgcn_sched_barrier(0);
      }
    }
    reduce_means();
  }

  // ======================= layers 1..6 (Cin = 36) =========================
  for (int layer = 1; layer < 7; ++layer) {
    const float* WeL = We + (size_t)(layer - 1) * HH * CINC;
    const float* WoL = Wo + (size_t)(layer - 1) * HH * CINC;

    // cooperative load of both weight slabs into LDS
    {
      float* dE = (float*)&sW2[0][0][0];
      float* dO = (float*)&sW2[1][0][0];
      for (int i = tid; i < HH * CINC; i += NTHREADS) {
        dE[i] = WeL[i];
        dO[i] = WoL[i];
      }
    }
    // effective biases: fold the n-uniform eg/og feature chunks
    if (tid < 2 * UU * HH) {
      const int e = tid >> 6;
      const int rem = tid & 63;
      const int u = rem >> 4, h = rem & 15;
      const float* WL = e ? WoL : WeL;
      const float* bL = e ? (bo + (size_t)(layer - 1) * HH)
                          : (be + (size_t)(layer - 1) * HH);
      float s = bL[h];
      #pragma unroll
      for (int c = 0; c < 8; ++c) {
        const float sum = sAccO[0][8 + c] + sAccO[1][8 + c] +
                          sAccO[2][8 + c] + sAccO[3][8 + c];
        const float eg = sAccE[u][8 + c] * invN;
        const float og = (sum - sAccO[u][8 + c]) * inv3 * invN;
        s += WL[h * CINC + 12 + c] * eg + WL[h * CINC + 28 + c] * og;
      }
      sBeff[e][u][h] = s;
    }
    __syncthreads();

    #pragma unroll
    for (int u = 0; u < UU; ++u) { vE[u] = 0.f; vO[u] = 0.f; }

    #pragma unroll
    for (int t = 0; t < TT; ++t) {
      const int n0 = (wave * TT + t) * 16;

      // 5-chunk conv1x1: chunk0 = f16 chA regs, chunks 1..4 = f16 LDS state;
      // weights & bias streamed from LDS too (minimal VGPR pressure).
      auto conv = [&](int u, int t_, int e) -> v8f {
        v8f d;
        const float bias = sBeff[e][u][p];
        #pragma unroll
        for (int k = 0; k < 8; ++k) d[k] = bias;
        {
          v2f a;
          a.x = (float)chA[u][t_].x;
          a.y = (float)chA[u][t_].y;
          d = wmma4(a, sW2[e][p][ksel >> 1], d);
        }
        const h2* rec = (const h2*)(fAh + ((size_t)u * NN + n0 + p) * REC_HALVES);
        #pragma unroll
        for (int q = 0; q < 4; ++q) {
          h2 hv = rec[q * 2 + (ksel >> 1)];
          v2f a;
          a.x = (float)hv.x;
          a.y = (float)hv.y;
          d = wmma4(a, sW2[e][p][(cbs[q] + ksel) >> 1], d);
        }
        return d;
      };

      // pass 1: opp for all users -> leave-one-out sum (wave-local)
      v8f os;
      #pragma unroll
      for (int k = 0; k < 8; ++k) os[k] = 0.f;
      #pragma unroll
      for (int u = 0; u < UU; ++u) {
        v8f o = conv(u, t, 1);
        #pragma unroll
        for (int k = 0; k < 8; ++k) {
          o[k] = fmaxf(o[k], 0.f);
          os[k] += o[k];
          vO[u] += o[k];
        }
        __builtin_amdgcn_sched_barrier(0);   // keep 1 accumulator live
      }
      // pass 2: ego + recomputed opp -> loo; store new locals into LDS state
      #pragma unroll
      for (int u = 0; u < UU; ++u) {
        v8f e = conv(u, t, 0);
        #pragma unroll
        for (int k = 0; k < 8; ++k) {
          e[k] = fmaxf(e[k], 0.f);
          vE[u] += e[k];
        }
        v8f o = conv(u, t, 1);
        v8f l;
        #pragma unroll
        for (int k = 0; k < 8; ++k) {
          o[k] = fmaxf(o[k], 0.f);
          l[k] = (os[k] - o[k]) * inv3;
        }
        xstore(e, u, n0, 0);
        xstore(l, u, n0, 8);
        __builtin_amdgcn_sched_barrier(0);
      }
    }
    reduce_means();
  }

  // ============== final conv (W8, Cout=1), mean over users, * pi ==========
  float uni = 0.f;
  {
    float sumO8[8];
    #pragma unroll
    for (int c = 0; c < 8; ++c) {
      float s = 0.f;
      #pragma unroll
      for (int u = 0; u < UU; ++u) s += sAccO[u][8 + c];
      sumO8[c] = s;
    }
    #pragma unroll
    for (int u = 0; u < UU; ++u)
      #pragma unroll
      for (int c = 0; c < 8; ++c) {
        uni += W8[12 + c] * (sAccE[u][8 + c] * invN);
        uni += W8[28 + c] * ((sumO8[c] - sAccO[u][8 + c]) * inv3 * invN);
      }
  }
  const float PI_F = 3.14159265358979323846f;
  #pragma unroll
  for (int t = 0; t < TT; ++t) {
    const int n0 = (wave * TT + t) * 16;
    float s = 0.f;
    #pragma unroll
    for (int u = 0; u < UU; ++u) {
      s += W8[0 + ksel] * (float)chA[u][t].x;
      s += W8[1 + ksel] * (float)chA[u][t].y;
      const h2* rec = (const h2*)(fAh + ((size_t)u * NN + n0 + p) * REC_HALVES);
      #pragma unroll
      for (int q = 0; q < 4; ++q) {
        h2 hv = rec[q * 2 + (ksel >> 1)];
        s += W8[cbs[q] + ksel]     * (float)hv.x;
        s += W8[cbs[q] + ksel + 1] * (float)hv.y;
      }
    }
    s += __shfl_xor(s, 16, 32);
    if (!hiH)
      out[(size_t)b * NN + n0 + p] = PI_F * (b8[0] + 0.25f * (s + uni));
    __builtin_amdgcn_sched_barrier(0);
  }
}

extern "C" void kernel_launch(void* const* d_in, const int* in_sizes, int n_in,
                              void* d_out, int out_size, void* d_ws, size_t ws_size,
                              hipStream_t stream) {
  const float* channel = (const float*)d_in[0];
  const float* We1 = (const float*)d_in[1];
  const float* be1 = (const float*)d_in[2];
  const float* We  = (const float*)d_in[3];
  const float* be  = (const float*)d_in[4];
  const float* Wo1 = (const float*)d_in[5];
  const float* bo1 = (const float*)d_in[6];
  const float* Wo  = (const float*)d_in[7];
  const float* bo  = (const float*)d_in[8];
  const float* W8  = (const float*)d_in[9];
  const float* b8  = (const float*)d_in[10];
  (void)d_ws; (void)ws_size; (void)n_in; (void)out_size;

  // allow >64KB dynamic LDS (gfx1250 WGP has 320 KB); host-side, idempotent,
  // not a stream op -> safe under graph capture.
  (void)hipFuncSetAttribute((const void*)risnet_pi_kernel,
                            hipFuncAttributeMaxDynamicSharedMemorySize,
                            FA_BYTES);

  const int Bn = in_sizes[0] / (UU * FF * NN);   // batch (64)
  risnet_pi_kernel<<<dim3(Bn), dim3(NTHREADS), FA_BYTES, stream>>>(
      channel, We1, be1, We, be, Wo1, bo1, Wo, bo, W8, b8, (float*)d_out);
}